// MultiScaleRetention_68685116998034
// MI455X (gfx1250) — compile-verified
//
#include <hip/hip_runtime.h>
#include <hip/hip_bf16.h>

typedef __attribute__((ext_vector_type(2))) float v2f;
typedef __attribute__((ext_vector_type(8))) float v8f;
typedef __attribute__((ext_vector_type(4))) int   v4i;
#define AS3 __attribute__((address_space(3)))

#define E_DIM 4096
#define B_DIM 64
#define H_DIM 16
#define DK_DIM 256
#define EPS_RMS 1e-6f

constexpr int MDIM   = 64;     // rows of activations (batch)
constexpr int KDIM   = 4096;   // reduction dim (compile-time -> immediate offsets)
constexpr int NDIM   = 4096;   // output columns
constexpr int KCHUNK = 32;     // K-slice staged in LDS per pipeline stage
constexpr int LDA    = 36;     // padded LDS row stride (dwords): 144B rows ->
                               // b128-aligned async stores + conflict-free ds_load_b64

// ---- CDNA5 async global->LDS path (guarded; falls back to ld/st copy) -------
#if __has_builtin(__builtin_amdgcn_s_wait_asynccnt)
#define WAIT_ASYNC(n) __builtin_amdgcn_s_wait_asynccnt(n)
#else
#define WAIT_ASYNC(n)
#endif

__device__ __forceinline__ void async_copy16(const float* __restrict__ g,
                                             float* __restrict__ l)
{
#if __has_builtin(__builtin_amdgcn_global_load_async_to_lds_b128)
    __builtin_amdgcn_global_load_async_to_lds_b128((v4i*)g, (AS3 v4i*)l, 0, 0);
#else
    *(float4*)l = *(const float4*)g;
#endif
}

__device__ __forceinline__ void stage_a_chunk(const float* __restrict__ A,
                                              float* __restrict__ tile,
                                              int kpos, int tid)
{
    // 64 rows x 32 dwords (8 KB) per chunk; 128 threads x 4 x b128.
    #pragma unroll
    for (int j = 0; j < 4; ++j) {
        const int idx = tid + 128 * j;
        const int row = idx >> 3;          // 0..63
        const int c   = (idx & 7) * 4;     // dword col 0,4,...,28
        async_copy16(A + (size_t)row * KDIM + kpos + c, tile + row * LDA + c);
    }
}

// ---------------------------------------------------------------------------
// C = A(64 x 4096) * B(4096 x 4096), fp32, V_WMMA_F32_16X16X4_F32.
// Block = 128 threads (4 waves). Wave w owns columns [n0+16w, n0+16w+16) and
// all 64 M rows (4 accumulators) -> weight stream B read ONCE per block.
// A k-slices double-buffered in LDS via async-to-LDS, shared by all 4 waves.
// B addresses: single moving base pointer, all strides immediate offsets.
// ---------------------------------------------------------------------------
__global__ __launch_bounds__(128)
void gemm64_wmma_f32(const float* __restrict__ A,
                     const float* __restrict__ B0, const float* __restrict__ B1,
                     const float* __restrict__ B2,
                     float* __restrict__ C0, float* __restrict__ C1,
                     float* __restrict__ C2,
                     int ksplit)
{
    const float* Bm = (blockIdx.y == 0) ? B0 : (blockIdx.y == 1 ? B1 : B2);
    float*       Cm = (blockIdx.y == 0) ? C0 : (blockIdx.y == 1 ? C1 : C2);
    Cm += (size_t)blockIdx.z * MDIM * NDIM;        // split-K partial buffer

    __shared__ __align__(16) float tile[2][MDIM * LDA];

    const int tid  = threadIdx.x;
    const int lane = tid & 31;
    const int wave = tid >> 5;
    const int half = lane >> 4;                    // 0: lanes 0-15, 1: 16-31
    const int l16  = lane & 15;
    const int ncol = blockIdx.x * 64 + wave * 16 + l16;

    const int kper    = KDIM / ksplit;
    const int kbeg    = blockIdx.z * kper;
    const int nchunks = kper / KCHUNK;

    stage_a_chunk(A, &tile[0][0], kbeg, tid);

    v8f acc[4] = {{}, {}, {}, {}};

    // Moving B pointer: starts at this lane's (row kbeg + 2*half, col ncol);
    // +NDIM (16 KB) and +KCHUNK*NDIM (512 KB) fit 24-bit immediates;
    // advances by the constant 4*NDIM per k-step.
    const float* bptr = Bm + ncol + (size_t)(kbeg + 2 * half) * NDIM;

    // Per-lane LDS base for A fragments; per-(m,kk) offsets are
    // compile-time immediates (m*16*LDA + kk) after unrolling.
    const float* ta = &tile[0][0] + l16 * LDA + 2 * half;

    for (int c = 0; c < nchunks; ++c) {
        if (c + 1 < nchunks) {
            stage_a_chunk(A, &tile[(c + 1) & 1][0], kbeg + (c + 1) * KCHUNK, tid);
            WAIT_ASYNC(4);     // previous chunk's 4 asyncs complete
        } else {
            WAIT_ASYNC(0);
        }
        __syncthreads();

        const float* tb = ta + (c & 1) * (MDIM * LDA);
        #pragma unroll
        for (int kk = 0; kk < KCHUNK; kk += 4) {
            // B fragment: VGPR0 = row kr, VGPR1 = row kr+1, col = ncol.
            v2f b;
            b.x = bptr[0];
            b.y = bptr[NDIM];
            __builtin_prefetch(bptr + KCHUNK * NDIM, 0, 1);
            #pragma unroll
            for (int m = 0; m < 4; ++m) {
                v2f a = *(const v2f*)(tb + m * 16 * LDA + kk);
                acc[m] = __builtin_amdgcn_wmma_f32_16x16x4_f32(
                             false, a, false, b, (short)0, acc[m], false, false);
            }
            bptr += 4 * NDIM;
        }
        __syncthreads();
    }

    // C layout: VGPR r -> M = 16m + r + 8*half, N = ncol.
    float* Cout = Cm + (size_t)(half * 8) * NDIM + ncol;
    #pragma unroll
    for (int m = 0; m < 4; ++m)
        #pragma unroll
        for (int r = 0; r < 8; ++r)
            Cout[(size_t)(m * 16 + r) * NDIM] = acc[m][r];
}

// ---------------------------------------------------------------------------
// Fused rotary + retention GEMV + RMS norm (pure-bandwidth: prev_kv streams
// through once).  out[b,h,d] = decay[h]*(prev_kv[b,h,d,:]·qr) + v[b,h,d]*(qr·kr)
// ---------------------------------------------------------------------------
__global__ __launch_bounds__(256)
void retention_rms(const float* __restrict__ q, const float* __restrict__ k,
                   const float* __restrict__ v,
                   const float* __restrict__ sinv, const float* __restrict__ cosv,
                   const float* __restrict__ decay,
                   const float* __restrict__ prev_kv,
                   float* __restrict__ out_norm)
{
    const int h = blockIdx.x;
    const int b = blockIdx.y;
    const int t = threadIdx.x;

    __shared__ float qr_s[DK_DIM];
    __shared__ float red[256];

    const int base = b * E_DIM + h * DK_DIM;

    float qv = q[base + t];
    float kv = k[base + t];
    float qn = (t & 1) ? q[base + t - 1] : -q[base + t + 1];
    float kn = (t & 1) ? k[base + t - 1] : -k[base + t + 1];
    float c  = cosv[t];
    float s  = sinv[t];
    float qr = qv * c + qn * s;
    float kr = kv * c + kn * s;

    qr_s[t] = qr;
    red[t]  = qr * kr;
    __syncthreads();
    for (int off = 128; off > 0; off >>= 1) {
        if (t < off) red[t] += red[t + off];
        __syncthreads();
    }
    const float s_qk = red[0];
    __syncthreads();

    const float4* row = (const float4*)(prev_kv +
        ((((size_t)b * H_DIM + h) * DK_DIM + t) * DK_DIM));
    float acc = 0.f;
    #pragma unroll 8
    for (int kk = 0; kk < DK_DIM / 4; ++kk) {
        float4 p = row[kk];
        const float* qp = qr_s + kk * 4;
        acc += p.x * qp[0] + p.y * qp[1] + p.z * qp[2] + p.w * qp[3];
    }
    float o = decay[h] * acc + v[base + t] * s_qk;

    red[t] = o * o;
    __syncthreads();
    for (int off = 128; off > 0; off >>= 1) {
        if (t < off) red[t] += red[t + off];
        __syncthreads();
    }
    const float ms = red[0] * (1.0f / 256.0f);
    o *= rsqrtf(ms + EPS_RMS);

    out_norm[base + t] = o;
}

// Deterministic split-K combine: out = p0 + p1
__global__ __launch_bounds__(256)
void add2(const float* __restrict__ a, const float* __restrict__ b,
          float* __restrict__ o, int n)
{
    const int i = blockIdx.x * 256 + threadIdx.x;
    if (i < n) o[i] = a[i] + b[i];
}

// ---------------------------------------------------------------------------
// Inputs: 0:x 1:sin 2:cos 3:decay 4:prev_kv 5:Wq 6:Wk 7:Wv 8:Wo
// Workspace: q | k | v | out_norm | p0 | p1   (6 x 64*4096 fp32 = 6 MB)
// ---------------------------------------------------------------------------
extern "C" void kernel_launch(void* const* d_in, const int* in_sizes, int n_in,
                              void* d_out, int out_size, void* d_ws, size_t ws_size,
                              hipStream_t stream)
{
    (void)in_sizes; (void)n_in; (void)out_size; (void)ws_size;

    const float* x      = (const float*)d_in[0];
    const float* sinv   = (const float*)d_in[1];
    const float* cosv   = (const float*)d_in[2];
    const float* decay  = (const float*)d_in[3];
    const float* prevkv = (const float*)d_in[4];
    const float* Wq     = (const float*)d_in[5];
    const float* Wk     = (const float*)d_in[6];
    const float* Wv     = (const float*)d_in[7];
    const float* Wo     = (const float*)d_in[8];
    float* out = (float*)d_out;

    float* ws = (float*)d_ws;
    const size_t mat = (size_t)B_DIM * E_DIM;
    float* qbuf = ws;
    float* kbuf = ws + mat;
    float* vbuf = ws + 2 * mat;
    float* nbuf = ws + 3 * mat;
    float* pbuf = ws + 4 * mat;     // 2 split-K partials (pbuf, pbuf + mat)

    // Phase 1: fused Q/K/V GEMMs (x @ {Wq,Wk,Wv}); 192 blocks, no split-K.
    dim3 gqkv(NDIM / 64, 3, 1);
    gemm64_wmma_f32<<<gqkv, 128, 0, stream>>>(x, Wq, Wk, Wv,
                                              qbuf, kbuf, vbuf, 1);

    // Phase 2: rotary + retention GEMV + RMS norm (prev_kv single-pass stream).
    dim3 gret(H_DIM, B_DIM);
    retention_rms<<<gret, 256, 0, stream>>>(qbuf, kbuf, vbuf, sinv, cosv,
                                            decay, prevkv, nbuf);

    // Phase 3: out @ Wo with split-K=2 for WGP coverage, then combine.
    dim3 gout(NDIM / 64, 1, 2);
    gemm64_wmma_f32<<<gout, 128, 0, stream>>>(nbuf, Wo, Wo, Wo,
                                              pbuf, pbuf, pbuf, 2);
    add2<<<(int)(mat / 256), 256, 0, stream>>>(pbuf, pbuf + mat, out, (int)mat);
}